// MixedSphericalGaussianDistribution_61254823575950
// MI455X (gfx1250) — compile-verified
//
#include <hip/hip_runtime.h>
#include <hip/hip_bf16.h>
#include <math.h>

typedef __attribute__((ext_vector_type(2))) float v2f;
typedef __attribute__((ext_vector_type(8))) float v8f;

#define LOG2E_F   1.44269504088896340736f
#define INV_2PI_F 0.15915494309189533577f
#define INV_4PI_F 0.07957747154594766788f

// D = A(mu: 16 comps x K4) * B(wi: K4 x 16 dirs).
// A layout (16x4 f32): lanes 0-15 hold M=lane, v0=K0(mu_x), v1=K1(mu_y);
//                      lanes 16-31 hold M=lane-16, v0=K2(mu_z), v1=K3(0).
// B layout (4x16 f32): lanes 0-15 hold N=lane, v0=K0(wi_x), v1=K1(wi_y);
//                      lanes 16-31 hold N=lane-16, v0=K2(wi_z), v1=K3(0).
// D layout (16x16 f32): VGPR v, lanes 0-15 -> (comp=v,    dir=lane),
//                              lanes 16-31 -> (comp=v+8,  dir=lane-16).
// => each lane's 8 D regs (x 4 groups) all belong to ONE direction, so the
//    64-component mixture sum is a per-lane FMA chain + one shfl_xor(16).
__global__ __launch_bounds__(256) void vmf_mix_wmma_kernel(
    const float* __restrict__ lambdas,
    const float* __restrict__ kappas,
    const float* __restrict__ thetas,
    const float* __restrict__ phis,
    const float* __restrict__ wi,
    float* __restrict__ out,
    int ndirs)
{
  const int  lane = threadIdx.x & 31;   // wave32
  const bool hi   = lane >= 16;
  const int  l16  = lane & 15;

  // ---- hoisted per-lane component parameters (loop-invariant) ----------
  v2f   A[4];            // mu tiles, one per group of 16 components
  float klog2e[4][8];    // kappa * log2(e) for comps this lane touches
  float wgt[4][8];       // lambda * vmf_norm(kappa)
#pragma unroll
  for (int g = 0; g < 4; ++g) {
    const int mcomp = g * 16 + l16;
    const float th = thetas[mcomp];
    const float ph = phis[mcomp];
    const float st  = __sinf(th);
    const float mux = st * __cosf(ph);
    const float muy = st * __sinf(ph);
    const float muz = __cosf(th);
    A[g].x = hi ? muz  : mux;
    A[g].y = hi ? 0.0f : muy;
#pragma unroll
    for (int v = 0; v < 8; ++v) {
      const int   c   = g * 16 + (hi ? 8 : 0) + v;
      const float kap = kappas[c];
      const float kk  = fmaxf(kap, 1e-8f);
      const float den = 1.0f - __builtin_amdgcn_exp2f(-2.0f * kk * LOG2E_F);
      const float nrm = (kap < 1e-5f) ? INV_4PI_F : (kk * INV_2PI_F) / den;
      wgt[g][v]    = lambdas[c] * nrm;
      klog2e[g][v] = kap * LOG2E_F;
    }
  }

  // hoisted lane-dependent address components (branch-free B loads)
  const int laneOfsA = 3 * l16 + (hi ? 2 : 0);  // wi_x (lo) / wi_z (hi)
  const int laneOfsB = 3 * l16 + 1;             // wi_y (always in range)

  const int ntiles = ndirs >> 4;                                 // 16 dirs/tile
  const int waveId = (int)((blockIdx.x * blockDim.x + threadIdx.x) >> 5);
  const int nwaves = (int)((gridDim.x * blockDim.x) >> 5);

  for (int tile = waveId; tile < ntiles; tile += nwaves) {
    const int tbase = tile * 48;                // 16 dirs * 3 floats
    const float bx = wi[tbase + laneOfsA];      // unconditional loads,
    const float by = wi[tbase + laneOfsB];      // no exec branching
    v2f B;
    B.x = bx;
    B.y = hi ? 0.0f : by;                       // v_cndmask, not a branch

    // 4 independent accumulators -> 4x ILP on the exp/fma chain
    float acc0 = 0.0f, acc1 = 0.0f, acc2 = 0.0f, acc3 = 0.0f;
#pragma unroll
    for (int g = 0; g < 4; ++g) {
      v8f c = {};
      v8f dm = __builtin_amdgcn_wmma_f32_16x16x4_f32(
          /*neg_a=*/false, A[g], /*neg_b=*/false, B,
          /*c_mod=*/(short)0, c, /*reuse_a=*/false, /*reuse_b=*/false);
      float a = 0.0f;
#pragma unroll
      for (int v = 0; v < 8; ++v) {
        // kappa*(dot-1) in log2 domain: fma(klog2e, dot, -klog2e)
        const float t = fmaf(klog2e[g][v], dm[v], -klog2e[g][v]);
        a = fmaf(wgt[g][v], __builtin_amdgcn_exp2f(t), a);
      }
      if (g == 0) acc0 = a; else if (g == 1) acc1 = a;
      else if (g == 2) acc2 = a; else acc3 = a;
    }
    const float acc = (acc0 + acc1) + (acc2 + acc3);

    // lanes 0-15 hold comps {16g+v}, lanes 16-31 hold comps {16g+8+v}
    const float tot = acc + __shfl_xor(acc, 16, 32);
    if (!hi) out[(tile << 4) + l16] = tot;      // coalesced 16-float store
  }
}

extern "C" void kernel_launch(void* const* d_in, const int* in_sizes, int n_in,
                              void* d_out, int out_size, void* d_ws, size_t ws_size,
                              hipStream_t stream) {
  const float* lambdas = (const float*)d_in[0];
  const float* kappas  = (const float*)d_in[1];
  const float* thetas  = (const float*)d_in[2];
  const float* phis    = (const float*)d_in[3];
  const float* wi      = (const float*)d_in[4];
  float* out = (float*)d_out;

  const int ndirs  = in_sizes[4] / 3;           // wi is [N,3]
  const int ntiles = ndirs >> 4;
  const int wavesPerBlock = 256 / 32;           // wave32: 8 waves / block
  int blocks = (ntiles + wavesPerBlock - 1) / wavesPerBlock;
  if (blocks > 8192) blocks = 8192;
  if (blocks < 1)    blocks = 1;

  vmf_mix_wmma_kernel<<<blocks, 256, 0, stream>>>(
      lambdas, kappas, thetas, phis, wi, out, ndirs);
}